// DCTPolicy_71279277244823
// MI455X (gfx1250) — compile-verified
//
#include <hip/hip_runtime.h>
#include <math.h>

// Geometry (fixed by the reference)
#define WW 4096          // W = 2^12
#define HH 4096          // H = 2^12
#define LOG2PI_D 1.8378770664093454835606594728112353

typedef float f32x4 __attribute__((ext_vector_type(4)));
typedef float v2f   __attribute__((ext_vector_type(2)));
typedef float v8f   __attribute__((ext_vector_type(8)));

// Per-row zigzag descriptor: byte v holds (k+1) for block position (u,v),
// 0 => coefficient not kept (output is zero).
// k map (from reference _zigzag_indices, first 16):
//   u=0: v0..v4 -> k 0,2,3,9,10   u=1: v0..v3 -> k 1,4,8,11
//   u=2: v0..v2 -> k 5,7,12       u=3: v0..v1 -> k 6,13
//   u=4: v0 -> 14                 u=5: v0 -> 15   u=6,7: none
__device__ __constant__ unsigned long long krow[8] = {
    0x0000000B0A040301ull,   // u = 0 : 1,3,4,10,11,0,0,0
    0x000000000C090502ull,   // u = 1 : 2,5,9,12
    0x00000000000D0806ull,   // u = 2 : 6,8,13
    0x0000000000000E07ull,   // u = 3 : 7,14
    0x000000000000000Full,   // u = 4 : 15
    0x0000000000000010ull,   // u = 5 : 16
    0x0000000000000000ull,   // u = 6
    0x0000000000000000ull    // u = 7
};

// One pass over the output image: gather sample (or 0) per element, coalesced
// nontemporal float4 stores. Each sample element is read exactly once.
__global__ void fill_coeffs(const float* __restrict__ mean,
                            const float* __restrict__ log_std,
                            const float* __restrict__ eps,
                            float* __restrict__ out, int total4) {
    int tid = blockIdx.x * blockDim.x + threadIdx.x;
    if (tid >= total4) return;
    int lin = tid << 2;                    // element index (multiple of 4)
    int x   = lin & (WW - 1);
    int y   = (lin >> 12) & (HH - 1);
    int c   = lin >> 24;                   // H*W = 2^24
    int u   = y & 7, v0 = x & 7;           // v0 in {0,4}
    int bh  = y >> 3, bw = x >> 3;
    // sample base: ((c*512 + bh)*512 + bw)*16
    int base = ((((c << 9) | bh) << 9) | bw) << 4;

    // u is wave-uniform: one scalar load of the packed row descriptor.
    int uu = __builtin_amdgcn_readfirstlane(u);
    unsigned long long row = krow[uu];

    f32x4 o;
    #pragma unroll
    for (int j = 0; j < 4; ++j) {
        int kk = (int)((row >> ((v0 + j) << 3)) & 0xFFull);   // k+1, 0 => zero
        float val = 0.0f;
        if (kk > 0) {
            int idx = base + kk - 1;
            val = fmaf(__expf(log_std[idx]), eps[idx], mean[idx]);
        }
        o[j] = val;
    }
    __builtin_nontemporal_store(o, (f32x4*)(out + lin));
}

// WMMA-based deterministic reduction: sum(eps^2) and sum(log_std).
// Each wave-iteration consumes one 128-float group per array with a single
// b128 load per lane (contiguous 512B per wave), feeding 16x16x4 f32 WMMAs
// with an all-ones B matrix: D = A*1 + C accumulates row sums (all columns
// of D identical). Branch-free; tail (< 128 elems) folded in by thread 0.
__global__ void reduce_wmma(const float* __restrict__ eps,
                            const float* __restrict__ log_std,
                            float* __restrict__ partials,
                            int ngroups, int n) {
    const int lane   = threadIdx.x & 31;
    const int wave   = (blockIdx.x * blockDim.x + threadIdx.x) >> 5;
    const int nwaves = (gridDim.x * blockDim.x) >> 5;

    v8f accE = {0.f,0.f,0.f,0.f,0.f,0.f,0.f,0.f};
    v8f accS = {0.f,0.f,0.f,0.f,0.f,0.f,0.f,0.f};
    v2f ones = {1.0f, 1.0f};

    const f32x4* __restrict__ e4 = (const f32x4*)eps;
    const f32x4* __restrict__ s4 = (const f32x4*)log_std;

    for (int g = wave; g < ngroups; g += nwaves) {     // wave-uniform loop
        int vi = (g << 5) + lane;                      // f32x4 index
        f32x4 ev = e4[vi];
        f32x4 sv = s4[vi];
        v2f ea0 = {ev[0] * ev[0], ev[1] * ev[1]};
        v2f ea1 = {ev[2] * ev[2], ev[3] * ev[3]};
        v2f sa0 = {sv[0], sv[1]};
        v2f sa1 = {sv[2], sv[3]};
        accE = __builtin_amdgcn_wmma_f32_16x16x4_f32(false, ea0, false, ones,
                                                     (short)0, accE, false, false);
        accS = __builtin_amdgcn_wmma_f32_16x16x4_f32(false, sa0, false, ones,
                                                     (short)0, accS, false, false);
        accE = __builtin_amdgcn_wmma_f32_16x16x4_f32(false, ea1, false, ones,
                                                     (short)0, accE, false, false);
        accS = __builtin_amdgcn_wmma_f32_16x16x4_f32(false, sa1, false, ones,
                                                     (short)0, accS, false, false);
    }

    // Per-lane: sum this lane's 8 accumulator rows. Columns of D are
    // identical, so total = column L over all 16 rows = pe(L) + pe(L+16).
    float pe = 0.f, ps = 0.f;
    #pragma unroll
    for (int r = 0; r < 8; ++r) { pe += accE[r]; ps += accS[r]; }
    float we = __shfl(pe, 0, 32) + __shfl(pe, 16, 32);
    float ws = __shfl(ps, 0, 32) + __shfl(ps, 16, 32);

    // Tail elements [ngroups*128, n): serial, deterministic, one thread.
    if (blockIdx.x == 0 && threadIdx.x == 0) {
        for (int i = ngroups << 7; i < n; ++i) {
            float e = eps[i];
            we += e * e;
            ws += log_std[i];
        }
    }

    __shared__ float se[8], ss[8];
    int wib = threadIdx.x >> 5;
    if ((threadIdx.x & 31) == 0) { se[wib] = we; ss[wib] = ws; }
    __syncthreads();
    if (threadIdx.x == 0) {
        float te = 0.f, ts = 0.f;
        #pragma unroll
        for (int i = 0; i < 8; ++i) { te += se[i]; ts += ss[i]; }  // fixed order
        partials[2 * blockIdx.x]     = te;
        partials[2 * blockIdx.x + 1] = ts;
    }
}

__global__ void finalize(const float* __restrict__ partials, int nparts,
                         float* __restrict__ out2, int n) {
    __shared__ float se[256], ss[256];
    int t = threadIdx.x;
    float te = 0.f, ts = 0.f;
    for (int i = t; i < nparts; i += 256) {           // fixed strided order
        te += partials[2 * i];
        ts += partials[2 * i + 1];
    }
    se[t] = te; ss[t] = ts;
    __syncthreads();
    for (int s = 128; s > 0; s >>= 1) {               // fixed-pairing tree
        if (t < s) { se[t] += se[t + s]; ss[t] += ss[t + s]; }
        __syncthreads();
    }
    if (t == 0) {
        double sumE = (double)se[0];                  // sum eps^2
        double sumS = (double)ss[0];                  // sum log_std
        double dn   = (double)n;
        double log_prob = -0.5 * (sumE + 2.0 * sumS + dn * LOG2PI_D);
        double entropy  = 0.5 * (1.0 + LOG2PI_D) * dn + sumS;
        out2[0] = (float)log_prob;
        out2[1] = (float)entropy;
    }
}

extern "C" void kernel_launch(void* const* d_in, const int* in_sizes, int n_in,
                              void* d_out, int out_size, void* d_ws, size_t ws_size,
                              hipStream_t stream) {
    const float* mean    = (const float*)d_in[0];
    const float* log_std = (const float*)d_in[1];
    const float* eps     = (const float*)d_in[2];
    // d_in[3] (flat_idx) intentionally unused: index map is derived analytically.
    float* out = (float*)d_out;

    int n      = in_sizes[0];          // 12,582,912
    int chw    = out_size - 2;         // 50,331,648
    int total4 = chw >> 2;

    fill_coeffs<<<(total4 + 255) / 256, 256, 0, stream>>>(mean, log_std, eps, out, total4);

    int G = 1024;                                          // reduction blocks
    while (G > 1 && (size_t)(2 * G) * sizeof(float) > ws_size) G >>= 1;
    int ngroups = n >> 7;                                  // 128-float groups
    float* partials = (float*)d_ws;

    reduce_wmma<<<G, 256, 0, stream>>>(eps, log_std, partials, ngroups, n);
    finalize<<<1, 256, 0, stream>>>(partials, G, out + chw, n);
}